// GraphSAGEForLinkPrediction_79096117723240
// MI455X (gfx1250) — compile-verified
//
#include <hip/hip_runtime.h>
#include <hip/hip_bf16.h>

#define IN_C   128
#define HID_C  256
#define OUT_C  128
#define NN     50000
#define NE     600000

typedef __attribute__((ext_vector_type(2))) float v2f;
typedef __attribute__((ext_vector_type(8))) float v8f;
typedef __attribute__((ext_vector_type(4))) unsigned v4u;
typedef __attribute__((ext_vector_type(8))) int v8i;
typedef __attribute__((ext_vector_type(4))) int v4i;

#if defined(__gfx1250__) && __has_builtin(__builtin_amdgcn_tensor_load_to_lds)
#define USE_TDM 1
#endif

// ---------------------------------------------------------------- utilities
__global__ void zero_kernel(float* __restrict__ p, int n) {
    int i = blockIdx.x * blockDim.x + threadIdx.x;
    if (i < n) p[i] = 0.0f;
}

__global__ void count_kernel(const long long* __restrict__ ei, float* __restrict__ cnt) {
    int e = blockIdx.x * blockDim.x + threadIdx.x;
    if (e < NE) {
        int dst = (int)ei[NE + e];
        atomicAdd(&cnt[dst], 1.0f);
    }
}

// One block per edge; lane c handles channel c. Reads coalesced from feat[src],
// atomic adds coalesced into agg[dst] (agg fits in the 192MB L2 -> RMW in L2).
template <int C>
__global__ void scatter_kernel(const float* __restrict__ feat,
                               const long long* __restrict__ ei,
                               float* __restrict__ agg) {
    int e = blockIdx.x;
    int c = threadIdx.x;          // blockDim.x == C
    int src = (int)ei[e];
    int dst = (int)ei[NE + e];
    atomicAdd(&agg[dst * C + c], feat[src * C + c]);
}

template <int C>
__global__ void mean_kernel(float* __restrict__ agg, const float* __restrict__ cnt) {
    int i = blockIdx.x * blockDim.x + threadIdx.x;   // over NN*C, exact multiple of 256
    int row = i / C;
    agg[i] *= (1.0f / fmaxf(cnt[row], 1.0f));
}

#ifdef USE_TDM
// Issue one TDM DMA: load a 16 x K_ fp32 tile (row-major, row stride K_ elems)
// starting at gbase into LDS at lds_byte, inserting 4 DWORDs of padding after
// every K_ DWORDs so the LDS row stride is (K_+4) floats (bank-conflict free).
// D# layout per CDNA5 ISA.
__device__ __forceinline__ void tdm_load_tile(const float* gbase, unsigned lds_byte,
                                              int K_, unsigned pad_interval_code) {
    unsigned long long ga = (unsigned long long)gbase;
    v4u g0;
    g0[0] = 1u;                                            // count=1 (valid), user mode
    g0[1] = lds_byte;                                      // lds_addr
    g0[2] = (unsigned)(ga & 0xffffffffu);                  // global_addr[31:0]
    g0[3] = (unsigned)((ga >> 32) & 0x01ffffffu)           // global_addr[56:32]
          | (2u << 30);                                    // type=2 ("image")
    v8i g1;
    g1[0] = (int)((2u << 16)                               // data_size=2 -> 4 bytes
                | (1u << 20)                               // pad_enable
                | (pad_interval_code << 22)                // 6 -> 128 DW, 7 -> 256 DW
                | (3u << 25));                             // pad_amount code 3 -> 4 DW
    g1[1] = (int)(((unsigned)K_ & 0xffffu) << 16);         // tensor_dim0[15:0]
    g1[2] = (int)(((unsigned)NN & 0xffffu) << 16);         // dim0[31:16]=0 | tensor_dim1[15:0]
    g1[3] = (int)(((unsigned)K_) << 16);                   // dim1[31:16]=0 | tile_dim0=K_
    g1[4] = 16;                                            // tile_dim1=16 | tile_dim2=0
    g1[5] = K_;                                            // tensor_dim0_stride[31:0]
    g1[6] = 0;                                             // stride0[47:32] | stride1[15:0]
    g1[7] = 0;                                             // tensor_dim1_stride[47:16]
    v4i gz4 = (v4i)0;                                      // groups 2/3 unused (2D tensor)
    v8i gz8 = (v8i)0;                                      // extra descriptor words (unused)
    __builtin_amdgcn_tensor_load_to_lds(g0, g1, gz4, gz4, gz8, 0);
}
#endif

// ---------------------------------------------------------------- WMMA GEMM
// Out[M=NN, N=LDW] = relu?( Aagg[M,K]*Wl[K,LDW] + Aroot[M,K]*Wr[K,LDW] + bias )
// Block: 256 threads = 8 waves. Each wave owns one 16x16 output tile;
// the block covers a 16(M) x 128(N) strip. A tiles staged in LDS (pad +4)
// by the Tensor Data Mover (one DMA per panel, issued by wave 0).
// No divergence in the compute section -> EXEC all-ones for every v_wmma.
//
// V_WMMA_F32_16X16X4_F32 fragment layout (ISA 7.12.2):
//   A 16x4 : lane<16 -> M=lane,   holds K=k+0,k+1 ; lane>=16 -> M=lane-16, K=k+2,k+3
//   B 4x16 : lane<16 -> N=lane,   holds K=k+0,k+1 ; lane>=16 -> N=lane-16, K=k+2,k+3
//   C/D    : vgpr r -> M = r + 8*(lane>=16), N = lane&15
template <int K, int LDW, int RELU>
__global__ void __launch_bounds__(256)
sage_gemm_kernel(const float* __restrict__ Aagg,
                 const float* __restrict__ Aroot,
                 const float* __restrict__ Wl,
                 const float* __restrict__ Wr,
                 const float* __restrict__ bias,
                 float* __restrict__ Out) {
    constexpr int PAD = 4;
    __shared__ float sAgg [16 * (K + PAD)];
    __shared__ float sRoot[16 * (K + PAD)];

    const int m_base = blockIdx.x * 16;
    const int wave   = threadIdx.x >> 5;
    const int lane   = threadIdx.x & 31;
    const int mrow   = lane & 15;              // A row / D column index base
    const int koff   = (lane >> 4) << 1;       // 0 (lanes 0-15) or 2 (lanes 16-31)
    const int n_base = blockIdx.y * 128 + wave * 16;
    const int ncol   = n_base + mrow;

#ifdef USE_TDM
    if (threadIdx.x < 32) {                    // wave 0 issues the two panel DMAs
        constexpr unsigned PI = (K == 128) ? 6u : 7u;   // pad every K DWORDs
        tdm_load_tile(Aagg  + (size_t)m_base * K, (unsigned)(unsigned long long)sAgg,  K, PI);
        tdm_load_tile(Aroot + (size_t)m_base * K, (unsigned)(unsigned long long)sRoot, K, PI);
        __builtin_amdgcn_s_wait_tensorcnt(0);  // issuing wave: DMAs complete
    }
    __syncthreads();                           // publish LDS to all 8 waves
#else
    // Cooperative fallback staging (host pass / non-TDM toolchains).
    for (int idx = threadIdx.x; idx < 16 * K; idx += 256) {
        int r = idx / K, c = idx % K;          // K is a power of two -> shifts
        sAgg [r * (K + PAD) + c] = Aagg [(m_base + r) * (size_t)K + c];
        sRoot[r * (K + PAD) + c] = Aroot[(m_base + r) * (size_t)K + c];
    }
    __syncthreads();
#endif

    v8f acc;
    {
        float bv = bias[ncol];                 // bias per output channel (lin_l only)
        #pragma unroll
        for (int r = 0; r < 8; ++r) acc[r] = bv;
    }

    const float* __restrict__ Arow0 = sAgg  + mrow * (K + PAD) + koff;
    const float* __restrict__ Arow1 = sRoot + mrow * (K + PAD) + koff;

    // Pass 1: mean-aggregated neighbors x lin_l weights
    #pragma unroll 4
    for (int k = 0; k < K; k += 4) {
        v2f a, b;
        a[0] = Arow0[k];
        a[1] = Arow0[k + 1];
        b[0] = Wl[(k + koff    ) * LDW + ncol];
        b[1] = Wl[(k + koff + 1) * LDW + ncol];
        acc = __builtin_amdgcn_wmma_f32_16x16x4_f32(false, a, false, b,
                                                    (short)0, acc, false, false);
    }
    // Pass 2: root features x lin_r weights (same accumulator)
    #pragma unroll 4
    for (int k = 0; k < K; k += 4) {
        v2f a, b;
        a[0] = Arow1[k];
        a[1] = Arow1[k + 1];
        b[0] = Wr[(k + koff    ) * LDW + ncol];
        b[1] = Wr[(k + koff + 1) * LDW + ncol];
        acc = __builtin_amdgcn_wmma_f32_16x16x4_f32(false, a, false, b,
                                                    (short)0, acc, false, false);
    }

    const int mstore = m_base + ((lane >> 4) << 3);   // +8 for upper half-wave
    #pragma unroll
    for (int r = 0; r < 8; ++r) {
        float v = acc[r];
        if (RELU) v = fmaxf(v, 0.0f);
        Out[(mstore + r) * LDW + ncol] = v;
    }
}

// ---------------------------------------------------------------- driver
extern "C" void kernel_launch(void* const* d_in, const int* in_sizes, int n_in,
                              void* d_out, int out_size, void* d_ws, size_t ws_size,
                              hipStream_t stream) {
    const float*     x    = (const float*)d_in[0];
    const long long* ei   = (const long long*)d_in[1];   // int64 [2, NE]
    const float*     W1l  = (const float*)d_in[2];       // [128,256]
    const float*     b1   = (const float*)d_in[3];       // [256]
    const float*     W1r  = (const float*)d_in[4];       // [128,256]
    const float*     W2l  = (const float*)d_in[5];       // [256,128]
    const float*     b2   = (const float*)d_in[6];       // [128]
    const float*     W2r  = (const float*)d_in[7];       // [256,128]
    float*           out  = (float*)d_out;               // [NN,128]

    // workspace layout (floats)
    float* ws   = (float*)d_ws;
    float* agg1 = ws;                              // NN*IN_C   = 6,400,000
    float* cnt  = agg1 + (size_t)NN * IN_C;        // NN        =    50,000
    float* h    = cnt  + NN;                       // NN*HID_C  = 12,800,000
    float* agg2 = h    + (size_t)NN * HID_C;       // NN*HID_C  = 12,800,000

    const int B = 256;

    // ---- layer 1 aggregation ----
    {   // zero agg1 + cnt (contiguous)
        int n = NN * IN_C + NN;
        zero_kernel<<<(n + B - 1) / B, B, 0, stream>>>(agg1, n);
    }
    count_kernel<<<(NE + B - 1) / B, B, 0, stream>>>(ei, cnt);
    scatter_kernel<IN_C><<<NE, IN_C, 0, stream>>>(x, ei, agg1);
    mean_kernel<IN_C><<<(NN * IN_C) / B, B, 0, stream>>>(agg1, cnt);

    // ---- layer 1 GEMM: h = relu(agg1*W1_l + x*W1_r + b1) ----
    {
        dim3 grid(NN / 16, HID_C / 128);           // 3125 x 2
        sage_gemm_kernel<IN_C, HID_C, 1><<<grid, B, 0, stream>>>(agg1, x, W1l, W1r, b1, h);
    }

    // ---- layer 2 aggregation ----
    {
        int n = NN * HID_C;
        zero_kernel<<<(n + B - 1) / B, B, 0, stream>>>(agg2, n);
    }
    scatter_kernel<HID_C><<<NE, HID_C, 0, stream>>>(h, ei, agg2);
    mean_kernel<HID_C><<<(NN * HID_C) / B, B, 0, stream>>>(agg2, cnt);

    // ---- layer 2 GEMM: out = agg2*W2_l + h*W2_r + b2 ----
    {
        dim3 grid(NN / 16, OUT_C / 128);           // 3125 x 1
        sage_gemm_kernel<HID_C, OUT_C, 0><<<grid, B, 0, stream>>>(agg2, h, W2l, W2r, b2, out);
    }
}